// SelfAttentionSpa_87454124081322
// MI455X (gfx1250) — compile-verified
//
#include <hip/hip_runtime.h>

// CDNA5 / gfx1250, wave32. fp32 WMMA 16x16x4.
typedef __attribute__((ext_vector_type(2))) float v2f;
typedef __attribute__((ext_vector_type(8))) float v8f;

#define CCH 120      // channels
#define NHW 4096     // H*W
#define NB  4        // batch
#define K4C 30       // 120 / K(=4) chunks
#define LST 132      // padded LDS row stride (floats) for 64-row fs tiles

__device__ __forceinline__ v8f wmma4(v2f a, v2f b, v8f c) {
  // D = A(16x4,f32) x B(4x16,f32) + C(16x16,f32)
  return __builtin_amdgcn_wmma_f32_16x16x4_f32(false, a, false, b, (short)0, c,
                                               false, false);
}
__device__ __forceinline__ float leaky(float x) { return x >= 0.f ? x : 0.01f * x; }

// ---------------------------------------------------------------------------
// 1x1 conv + bias + LeakyReLU as GEMM: Y[o,hw] = leaky(sum_i W[o,i]*(X+Add)[i,hw]+B[o])
// Layouts: X/Add/Y are per-batch flat [C][HW] (== raw-reshape fs layout).
// Grid: (HW/16, B), block 256 (8 waves). Wave w owns output rows 16w..16w+15.
// ---------------------------------------------------------------------------
__global__ __launch_bounds__(256)
void conv1x1_leaky_kernel(const float* __restrict__ X,
                          const float* __restrict__ Add,
                          const float* __restrict__ Wm,
                          const float* __restrict__ Bias,
                          float* __restrict__ Y) {
  __shared__ float xt[CCH][16];
  const int b   = blockIdx.y;
  const int hw0 = blockIdx.x * 16;
  const int tid = threadIdx.x;
  const int wid = tid >> 5;
  const int lane = tid & 31;
  const int lh = lane & 15;
  const int hi = lane >> 4;
  const float* Xb = X + (size_t)b * CCH * NHW;
  const float* Ab = Add ? Add + (size_t)b * CCH * NHW : nullptr;

  for (int idx = tid; idx < CCH * 16; idx += 256) {
    int i = idx >> 4, j = idx & 15;
    float v = Xb[i * NHW + hw0 + j];
    if (Ab) v += Ab[i * NHW + hw0 + j];
    xt[i][j] = v;
  }
  __syncthreads();

  // A fragments: 16 rows of Wm held in VGPRs (lane L -> row obase+L).
  const int obase = wid * 16;
  const int orow  = (obase + lh < CCH) ? (obase + lh) : (CCH - 1);
  v2f areg[K4C];
#pragma unroll
  for (int kc = 0; kc < K4C; ++kc) {
    int k0 = 4 * kc + 2 * hi;
    areg[kc].x = Wm[orow * CCH + k0];
    areg[kc].y = Wm[orow * CCH + k0 + 1];
  }

  v8f acc = {};
#pragma unroll
  for (int kc = 0; kc < K4C; ++kc) {
    int k0 = 4 * kc + 2 * hi;
    v2f bf;
    bf.x = xt[k0][lh];
    bf.y = xt[k0 + 1][lh];
    acc = wmma4(areg[kc], bf, acc);
  }

  float* Yb = Y + (size_t)b * CCH * NHW;
#pragma unroll
  for (int r = 0; r < 8; ++r) {
    int o = obase + r + 8 * hi;
    if (o < CCH)
      Yb[o * NHW + hw0 + lh] = leaky(acc[r] + Bias[o]);
  }
}

// ---------------------------------------------------------------------------
// Pass A: column softmax stats of S = fs @ fs^T. S symmetric => column stats
// over n of column m == row stats of row m.  Wave owns 16 m-rows (A-fragments
// in registers), streams all n through LDS. Online (max, expsum) per row,
// then a 16-lane shuffle reduction.
// Grid: (NHW/128, B), block 256.
// ---------------------------------------------------------------------------
__global__ __launch_bounds__(256)
void attn_stats_kernel(const float* __restrict__ G,
                       float* __restrict__ Mst,
                       float* __restrict__ Zst) {
  __shared__ float gt[64 * LST];
  const int b = blockIdx.y;
  const int tid = threadIdx.x;
  const int wid = tid >> 5;
  const int lane = tid & 31;
  const int lh = lane & 15;
  const int hi = lane >> 4;
  const int mbase = blockIdx.x * 128 + wid * 16;
  const float* Gb = G + (size_t)b * NHW * CCH;   // fs as [NHW][CCH]

  v2f areg[K4C];
#pragma unroll
  for (int kc = 0; kc < K4C; ++kc) {
    int k0 = 4 * kc + 2 * hi;
    areg[kc].x = Gb[(mbase + lh) * CCH + k0];
    areg[kc].y = Gb[(mbase + lh) * CCH + k0 + 1];
  }

  float runM[8], runZ[8];
#pragma unroll
  for (int r = 0; r < 8; ++r) { runM[r] = -__builtin_inff(); runZ[r] = 0.f; }

  for (int n0 = 0; n0 < NHW; n0 += 64) {
    __syncthreads();
    for (int idx = tid; idx < 64 * CCH; idx += 256) {
      int r = idx / CCH, c = idx - r * CCH;
      gt[r * LST + c] = Gb[(n0 + r) * CCH + c];
    }
    __syncthreads();
#pragma unroll 1
    for (int s = 0; s < 4; ++s) {
      const float* row = &gt[(s * 16 + lh) * LST];   // B[K,N]: N=lh
      v8f sacc = {};
#pragma unroll
      for (int kc = 0; kc < K4C; ++kc) {
        int k0 = 4 * kc + 2 * hi;
        v2f bf; bf.x = row[k0]; bf.y = row[k0 + 1];
        sacc = wmma4(areg[kc], bf, sacc);
      }
#pragma unroll
      for (int r = 0; r < 8; ++r) {
        float v  = sacc[r];
        float nm = fmaxf(runM[r], v);
        runZ[r]  = runZ[r] * __expf(runM[r] - nm) + __expf(v - nm);
        runM[r]  = nm;
      }
    }
  }
  // combine the 16 lanes of each half-wave (each lane saw n == lh mod 16)
#pragma unroll
  for (int r = 0; r < 8; ++r) {
    float m = runM[r], z = runZ[r];
#pragma unroll
    for (int off = 1; off < 16; off <<= 1) {
      float om = __shfl_xor(m, off, 32);
      float oz = __shfl_xor(z, off, 32);
      float nm = fmaxf(m, om);
      z = z * __expf(m - nm) + oz * __expf(om - nm);
      m = nm;
    }
    if (lh == 0) {
      int mrow = mbase + r + 8 * hi;
      Mst[b * NHW + mrow] = m;
      Zst[b * NHW + mrow] = z;
    }
  }
}

// ---------------------------------------------------------------------------
// Pass B: out[n,c] = sum_m exp(S[n,m]-M[m])/Z[m] * fs[m,c].
// Wave owns 16 n-rows (B-fragments in registers). For each streamed 16-m tile:
//   S'^T tile (m x n) via WMMA, p^T = exp(.)*rZ, round-trip p^T through a
//   per-wave LDS patch to re-fragment, then out^T[c,n] += fs_tile^T x p^T.
// Grid: (NHW/128, B), block 256.
// ---------------------------------------------------------------------------
__global__ __launch_bounds__(256)
void attn_out_kernel(const float* __restrict__ G,
                     const float* __restrict__ Mst,
                     const float* __restrict__ Zst,
                     float* __restrict__ O) {
  __shared__ float gt[64 * LST];
  __shared__ float mrow_s[64];
  __shared__ float zrow_s[64];
  __shared__ float pbuf[8][16 * 17];   // per-wave P^T patch [m][n], stride 17
  const int b = blockIdx.y;
  const int tid = threadIdx.x;
  const int wid = tid >> 5;
  const int lane = tid & 31;
  const int lh = lane & 15;
  const int hi = lane >> 4;
  const int nbase = blockIdx.x * 128 + wid * 16;
  const float* Gb = G + (size_t)b * NHW * CCH;

  // zero LDS pad columns once (cols 120..131 stay 0; masked outputs only)
  for (int idx = tid; idx < 64 * LST; idx += 256) gt[idx] = 0.f;

  v2f breg[K4C];                       // this wave's 16 n-rows of fs
#pragma unroll
  for (int kc = 0; kc < K4C; ++kc) {
    int k0 = 4 * kc + 2 * hi;
    breg[kc].x = Gb[(nbase + lh) * CCH + k0];
    breg[kc].y = Gb[(nbase + lh) * CCH + k0 + 1];
  }

  v8f oacc[8];
#pragma unroll
  for (int cc = 0; cc < 8; ++cc) oacc[cc] = (v8f){};

  float* pw = &pbuf[wid][0];

  for (int m0 = 0; m0 < NHW; m0 += 64) {
    __syncthreads();
    for (int idx = tid; idx < 64 * CCH; idx += 256) {
      int r = idx / CCH, c = idx - r * CCH;
      gt[r * LST + c] = Gb[(m0 + r) * CCH + c];
    }
    for (int idx = tid; idx < 64; idx += 256) {
      mrow_s[idx] = Mst[b * NHW + m0 + idx];
      zrow_s[idx] = 1.0f / Zst[b * NHW + m0 + idx];
    }
    __syncthreads();
#pragma unroll 1
    for (int s = 0; s < 4; ++s) {
      // S'^T tile: A = fs m-rows (from LDS), B = fs n-rows (registers)
      const float* arow = &gt[(s * 16 + lh) * LST];
      v8f sacc = {};
#pragma unroll
      for (int kc = 0; kc < K4C; ++kc) {
        int k0 = 4 * kc + 2 * hi;
        v2f af; af.x = arow[k0]; af.y = arow[k0 + 1];
        sacc = wmma4(af, breg[kc], sacc);
      }
      // p^T[m][n] into per-wave LDS patch
#pragma unroll
      for (int r = 0; r < 8; ++r) {
        int ml = s * 16 + r + 8 * hi;                 // m local to 64-chunk
        float p = __expf(sacc[r] - mrow_s[ml]) * zrow_s[ml];
        pw[(r + 8 * hi) * 17 + lh] = p;
      }
      __asm volatile("s_wait_dscnt 0" ::: "memory");  // wave-internal LDS RAW
      // re-fragment p^T as B operands (K = the 16 m's of this subtile)
      v2f b2[4];
#pragma unroll
      for (int kc2 = 0; kc2 < 4; ++kc2) {
        int k0 = 4 * kc2 + 2 * hi;
        b2[kc2].x = pw[k0 * 17 + lh];
        b2[kc2].y = pw[(k0 + 1) * 17 + lh];
      }
      // out^T[c,n] += fs_tile^T (c x m) @ p^T (m x n)
#pragma unroll
      for (int cc = 0; cc < 8; ++cc) {
        int cb = cc * 16;
#pragma unroll
        for (int kc2 = 0; kc2 < 4; ++kc2) {
          int mk = s * 16 + 4 * kc2 + 2 * hi;
          v2f a2;
          a2.x = gt[mk * LST + cb + lh];
          a2.y = gt[(mk + 1) * LST + cb + lh];
          oacc[cc] = wmma4(a2, b2[kc2], oacc[cc]);
        }
      }
    }
  }
  // oacc[cc][r] = out[n = nbase+lh][c = 16cc + r + 8hi]
  float* Ob = O + (size_t)b * NHW * CCH;
#pragma unroll
  for (int cc = 0; cc < 8; ++cc) {
#pragma unroll
    for (int r = 0; r < 8; ++r) {
      int c = cc * 16 + r + 8 * hi;
      if (c < CCH)
        Ob[(nbase + lh) * CCH + c] = oacc[cc][r];
    }
  }
}

// ---------------------------------------------------------------------------
extern "C" void kernel_launch(void* const* d_in, const int* in_sizes, int n_in,
                              void* d_out, int out_size, void* d_ws, size_t ws_size,
                              hipStream_t stream) {
  const float* feature = (const float*)d_in[0];
  const float* w1 = (const float*)d_in[1];
  const float* b1 = (const float*)d_in[2];
  const float* w2 = (const float*)d_in[3];
  const float* b2 = (const float*)d_in[4];
  float* out = (float*)d_out;

  // workspace: fs (7.9MB) | attn-out (7.9MB) | M (64KB) | Z (64KB)
  float* G   = (float*)d_ws;
  float* O   = G + (size_t)NB * NHW * CCH;
  float* Mst = O + (size_t)NB * NHW * CCH;
  float* Zst = Mst + (size_t)NB * NHW;

  dim3 blk(256);
  conv1x1_leaky_kernel<<<dim3(NHW / 16, NB), blk, 0, stream>>>(feature, nullptr, w1, b1, G);
  attn_stats_kernel  <<<dim3(NHW / 128, NB), blk, 0, stream>>>(G, Mst, Zst);
  attn_out_kernel    <<<dim3(NHW / 128, NB), blk, 0, stream>>>(G, Mst, Zst, O);
  conv1x1_leaky_kernel<<<dim3(NHW / 16, NB), blk, 0, stream>>>(feature, O, w2, b2, out);
}